// SAttention_without_softmax_42769284334295
// MI455X (gfx1250) — compile-verified
//
#include <hip/hip_runtime.h>
#include <cstddef>
#include <cstdint>

typedef __attribute__((ext_vector_type(2))) float v2f;
typedef __attribute__((ext_vector_type(8))) float v8f;

namespace sattn {

constexpr int T = 4, B = 8, N = 196, C = 768, H = 12, D = 64;
constexpr int C3 = 3 * C;            // 2304
constexpr int BN = B * N;            // 1568 (= 49 * 32, no M edge)
constexpr int BH = B * H;            // 96
constexpr int TN2 = 7;               // ceil(196/32): 32-wide tiles, edges clamped
constexpr float THETA   = 1.0f;
constexpr float POSLIM  = 1.0f;      // LEVEL - 1
constexpr float PREFIRE_V = 0.2f;    // PREFIRE * THETA
constexpr float SCALE   = 0.125f;    // D^-0.5
constexpr float INV_N   = 1.0f / (float)N;

constexpr size_t SZ_HD  = (size_t)B * H * N * D;  // 1,204,224
constexpr size_t SZ_NN  = (size_t)B * H * N * N;  // 3,687,936
constexpr size_t SZ_QKV = (size_t)B * N * C3;     // 3,612,672

// workspace layout (float offsets); vq..aa contiguous for one-shot init
constexpr size_t O_VQ   = 0;
constexpr size_t O_AQ   = O_VQ + SZ_HD;
constexpr size_t O_VK   = O_AQ + SZ_HD;
constexpr size_t O_AK   = O_VK + SZ_HD;
constexpr size_t O_VV   = O_AK + SZ_HD;
constexpr size_t O_AV   = O_VV + SZ_HD;
constexpr size_t O_VO   = O_AV + SZ_HD;
constexpr size_t O_AO   = O_VO + SZ_HD;
constexpr size_t O_VA   = O_AO + SZ_HD;   // attn membrane (prefired)
constexpr size_t O_AA   = O_VA + SZ_NN;
constexpr size_t O_QKV  = O_AA + SZ_NN;
constexpr size_t O_A1   = O_QKV + SZ_QKV; // q + q_sum   (= aq*scale)
constexpr size_t O_A2   = O_A1 + SZ_HD;   // q           (= spike*scale)
constexpr size_t O_B1   = O_A2 + SZ_HD;   // k
constexpr size_t O_B2   = O_B1 + SZ_HD;   // k_sum
constexpr size_t O_V1   = O_B2 + SZ_HD;   // v
constexpr size_t O_V2   = O_V1 + SZ_HD;   // v_sum
constexpr size_t O_ATTN = O_V2 + SZ_HD;   // attn_raw; later reused as y_pre
constexpr size_t O_C1   = O_ATTN + SZ_NN; // attn + attn_sum (= aa)
constexpr size_t O_C2   = O_C1 + SZ_NN;   // attn            (= spike)
constexpr size_t O_Y2   = O_C2 + SZ_NN;   // [B,N,C] post-IF, head-merged

__device__ __forceinline__ int imin(int a, int b) { return a < b ? a : b; }

// ---------------- WMMA helper (V_WMMA_F32_16X16X4_F32) --------------------
__device__ __forceinline__ v8f wmma4(v2f a, v2f b, v8f c) {
  return __builtin_amdgcn_wmma_f32_16x16x4_f32(false, a, false, b,
                                               (short)0, c, false, false);
}
__device__ __forceinline__ v2f ld2(const float* p) {       // const-offset b64
  return *(const v2f*)p;
}
__device__ __forceinline__ v2f ld2col(const float* p, int ldb) {
  v2f r; r.x = p[0]; r.y = p[ldb]; return r;
}

// ---------------- kernels ------------------------------------------------
__global__ void k_init(float* __restrict__ ws) {
  size_t i = (size_t)blockIdx.x * blockDim.x + threadIdx.x;
  size_t nz = 8 * SZ_HD;
  size_t total = nz + 2 * SZ_NN;
  if (i >= total) return;
  ws[i] = (i >= nz && i < nz + SZ_NN) ? PREFIRE_V : 0.0f;  // O_VA prefired
}

// C = A(MxK) @ B(KxN) [+ bias]; dims multiples of 32, K multiple of 32.
// One wave per 32x32 tile (2x2 WMMA blocking); group-of-8 k-steps so all
// in-group load offsets are immediates; 3 pointer bumps per 32 WMMAs.
template <int Mdim, int Ndim, int Kdim, int LDA, int LDB, int LDC, bool BIAS>
__global__ __launch_bounds__(256) void k_gemm_nn(
    const float* __restrict__ A, const float* __restrict__ Bm,
    float* __restrict__ Cp, const float* __restrict__ bias) {
  constexpr int tM = Mdim / 32, tN = Ndim / 32;
  int gwave = (int)((blockIdx.x * blockDim.x + threadIdx.x) >> 5);
  int lane  = threadIdx.x & 31;
  if (gwave >= tM * tN) return;
  int tm = gwave / tN, tn = gwave - tm * tN;
  int l15 = lane & 15, hh = lane >> 4, kh = 2 * hh;

  const float* pA0 = A + (uint32_t)(tm * 32 + l15) * LDA + kh;
  const float* pA1 = pA0 + 16u * LDA;
  const float* pB  = Bm + (uint32_t)kh * LDB + tn * 32 + l15;

  v8f acc00 = {}, acc01 = {}, acc10 = {}, acc11 = {};
  for (int g = 0; g < Kdim / 32; ++g) {   // 8 k-steps of 4 per group
#pragma unroll
    for (int u = 0; u < 8; ++u) {
      v2f A0 = ld2(pA0 + 4 * u);
      v2f A1 = ld2(pA1 + 4 * u);
      v2f B0 = ld2col(pB + 4 * u * LDB, LDB);
      v2f B1 = ld2col(pB + 4 * u * LDB + 16, LDB);
      acc00 = wmma4(A0, B0, acc00);
      acc01 = wmma4(A0, B1, acc01);
      acc10 = wmma4(A1, B0, acc10);
      acc11 = wmma4(A1, B1, acc11);
    }
    pA0 += 32; pA1 += 32; pB += 32 * LDB;
  }
  uint32_t n0 = tn * 32 + l15, n1 = n0 + 16;
  float bv0 = BIAS ? bias[n0] : 0.0f;
  float bv1 = BIAS ? bias[n1] : 0.0f;
  float* pC = Cp + (uint32_t)(tm * 32 + 8 * hh) * LDC;
#pragma unroll
  for (int v = 0; v < 8; ++v) {
    pC[(uint32_t)v * LDC + n0]        = acc00[v] + bv0;
    pC[(uint32_t)v * LDC + n1]        = acc01[v] + bv1;
    pC[(uint32_t)(16 + v) * LDC + n0] = acc10[v] + bv0;
    pC[(uint32_t)(16 + v) * LDC + n1] = acc11[v] + bv1;
  }
}

// attn_raw[bh] = A1 @ B1^T + A2 @ B2^T  (per-(b,h), K = D = 64).
// 32x32 per wave; edge lanes clamp their row/col (garbage stays in their own
// masked-out row/col of D). K loop fully unrolled: zero in-loop address ALU.
__global__ __launch_bounds__(256) void k_attn(
    const float* __restrict__ A1, const float* __restrict__ A2,
    const float* __restrict__ B1, const float* __restrict__ B2,
    float* __restrict__ Out) {
  constexpr int TILES = TN2 * TN2;  // 49
  int gwave = (int)((blockIdx.x * blockDim.x + threadIdx.x) >> 5);
  int lane  = threadIdx.x & 31;
  if (gwave >= BH * TILES) return;
  int bh = gwave / TILES, tidx = gwave - bh * TILES;
  int tm = tidx / TN2, tn = tidx - tm * TN2;
  int l15 = lane & 15, hh = lane >> 4, kh = 2 * hh;
  uint32_t off = (uint32_t)bh * (N * D);

  uint32_t am0 = off + (uint32_t)imin(tm * 32 + l15, N - 1) * D + kh;
  uint32_t am1 = off + (uint32_t)imin(tm * 32 + 16 + l15, N - 1) * D + kh;
  uint32_t bn0 = off + (uint32_t)imin(tn * 32 + l15, N - 1) * D + kh;
  uint32_t bn1 = off + (uint32_t)imin(tn * 32 + 16 + l15, N - 1) * D + kh;
  const float *pA1m0 = A1 + am0, *pA1m1 = A1 + am1;
  const float *pA2m0 = A2 + am0, *pA2m1 = A2 + am1;
  const float *pB1n0 = B1 + bn0, *pB1n1 = B1 + bn1;
  const float *pB2n0 = B2 + bn0, *pB2n1 = B2 + bn1;

  v8f acc00 = {}, acc01 = {}, acc10 = {}, acc11 = {};
#pragma unroll
  for (int k = 0; k < D / 4; ++k) {       // 16 steps, fully unrolled
    v2f a1m0 = ld2(pA1m0 + 4 * k), a1m1 = ld2(pA1m1 + 4 * k);
    v2f a2m0 = ld2(pA2m0 + 4 * k), a2m1 = ld2(pA2m1 + 4 * k);
    v2f b1n0 = ld2(pB1n0 + 4 * k), b1n1 = ld2(pB1n1 + 4 * k);
    v2f b2n0 = ld2(pB2n0 + 4 * k), b2n1 = ld2(pB2n1 + 4 * k);
    acc00 = wmma4(a1m0, b1n0, acc00); acc00 = wmma4(a2m0, b2n0, acc00);
    acc01 = wmma4(a1m0, b1n1, acc01); acc01 = wmma4(a2m0, b2n1, acc01);
    acc10 = wmma4(a1m1, b1n0, acc10); acc10 = wmma4(a2m1, b2n0, acc10);
    acc11 = wmma4(a1m1, b1n1, acc11); acc11 = wmma4(a2m1, b2n1, acc11);
  }
  uint32_t obase = (uint32_t)bh * (N * N);
  int cn0 = tn * 32 + l15, cn1 = cn0 + 16;
  int mr0 = tm * 32 + 8 * hh, mr1 = mr0 + 16;
  bool okn0 = cn0 < N, okn1 = cn1 < N;
#pragma unroll
  for (int v = 0; v < 8; ++v) {
    if (mr0 + v < N) {
      if (okn0) Out[obase + (uint32_t)(mr0 + v) * N + cn0] = acc00[v];
      if (okn1) Out[obase + (uint32_t)(mr0 + v) * N + cn1] = acc01[v];
    }
    if (mr1 + v < N) {
      if (okn0) Out[obase + (uint32_t)(mr1 + v) * N + cn0] = acc10[v];
      if (okn1) Out[obase + (uint32_t)(mr1 + v) * N + cn1] = acc11[v];
    }
  }
}

// y_pre[bh] = C1 @ V1 + C2 @ V2  (per-(b,h), K = N = 196 = 7 groups x 7 steps)
__global__ __launch_bounds__(256) void k_yg(
    const float* __restrict__ C1, const float* __restrict__ C2,
    const float* __restrict__ V1, const float* __restrict__ V2,
    float* __restrict__ Out) {
  constexpr int TND = D / 32;          // 2
  constexpr int TILES = TN2 * TND;     // 14
  int gwave = (int)((blockIdx.x * blockDim.x + threadIdx.x) >> 5);
  int lane  = threadIdx.x & 31;
  if (gwave >= BH * TILES) return;
  int bh = gwave / TILES, tidx = gwave - bh * TILES;
  int tm = tidx / TND, tn = tidx - tm * TND;
  int l15 = lane & 15, hh = lane >> 4, kh = 2 * hh;

  uint32_t offA = (uint32_t)bh * (N * N);
  uint32_t offB = (uint32_t)bh * (N * D);
  uint32_t am0 = offA + (uint32_t)imin(tm * 32 + l15, N - 1) * N + kh;
  uint32_t am1 = offA + (uint32_t)imin(tm * 32 + 16 + l15, N - 1) * N + kh;
  const float *pC1m0 = C1 + am0, *pC1m1 = C1 + am1;
  const float *pC2m0 = C2 + am0, *pC2m1 = C2 + am1;
  uint32_t b0 = offB + (uint32_t)kh * D + tn * 32 + l15;
  const float *pV1 = V1 + b0, *pV2 = V2 + b0;

  v8f acc00 = {}, acc01 = {}, acc10 = {}, acc11 = {};
  for (int g = 0; g < 7; ++g) {          // 7 groups of 7 k-steps (K=196)
#pragma unroll
    for (int u = 0; u < 7; ++u) {
      v2f c1m0 = ld2(pC1m0 + 4 * u), c1m1 = ld2(pC1m1 + 4 * u);
      v2f c2m0 = ld2(pC2m0 + 4 * u), c2m1 = ld2(pC2m1 + 4 * u);
      v2f v1n0 = ld2col(pV1 + 4 * u * D, D);
      v2f v1n1 = ld2col(pV1 + 4 * u * D + 16, D);
      v2f v2n0 = ld2col(pV2 + 4 * u * D, D);
      v2f v2n1 = ld2col(pV2 + 4 * u * D + 16, D);
      acc00 = wmma4(c1m0, v1n0, acc00); acc00 = wmma4(c2m0, v2n0, acc00);
      acc01 = wmma4(c1m0, v1n1, acc01); acc01 = wmma4(c2m0, v2n1, acc01);
      acc10 = wmma4(c1m1, v1n0, acc10); acc10 = wmma4(c2m1, v2n0, acc10);
      acc11 = wmma4(c1m1, v1n1, acc11); acc11 = wmma4(c2m1, v2n1, acc11);
    }
    pC1m0 += 28; pC1m1 += 28; pC2m0 += 28; pC2m1 += 28;
    pV1 += 28 * D; pV2 += 28 * D;
  }
  uint32_t n0 = tn * 32 + l15, n1 = n0 + 16;
  int mr0 = tm * 32 + 8 * hh, mr1 = mr0 + 16;
#pragma unroll
  for (int v = 0; v < 8; ++v) {
    if (mr0 + v < N) {
      Out[offB + (uint32_t)(mr0 + v) * D + n0] = acc00[v];
      Out[offB + (uint32_t)(mr0 + v) * D + n1] = acc01[v];
    }
    if (mr1 + v < N) {
      Out[offB + (uint32_t)(mr1 + v) * D + n0] = acc10[v];
      Out[offB + (uint32_t)(mr1 + v) * D + n1] = acc11[v];
    }
  }
}

// IF over q/k/v + build attention operands. i indexes [B,H,N,D].
__global__ void k_if_qkv(const float* __restrict__ qkv,
                         float* vq, float* aq, float* vk, float* ak,
                         float* vv, float* av,
                         float* A1, float* A2, float* B1, float* B2,
                         float* V1, float* V2) {
  size_t i = (size_t)blockIdx.x * blockDim.x + threadIdx.x;
  if (i >= SZ_HD) return;
  int d = (int)(i % D); size_t r = i / D;
  int n = (int)(r % N); r /= N;
  int h = (int)(r % H); int b = (int)(r / H);
  size_t base = ((size_t)b * N + n) * C3 + (size_t)h * D + d;
  float xq = qkv[base], xk = qkv[base + C], xv = qkv[base + 2 * C];
  { // q (asymmetric)
    float v = vq[i] + xq, a = aq[i];
    float s = (v >= THETA && a < POSLIM) ? 1.f
            : ((v < 0.f && a > 0.f) ? -1.f : 0.f);
    vq[i] = v - s * THETA; a += s; aq[i] = a;
    A1[i] = a * (SCALE * THETA);   // q + q_sum
    A2[i] = s * (SCALE * THETA);   // q
  }
  { // k (asymmetric)
    float v = vk[i] + xk, a = ak[i];
    float s = (v >= THETA && a < POSLIM) ? 1.f
            : ((v < 0.f && a > 0.f) ? -1.f : 0.f);
    vk[i] = v - s * THETA; a += s; ak[i] = a;
    B1[i] = s * THETA;             // k
    B2[i] = (a - s) * THETA;       // k_sum
  }
  { // v (symmetric)
    float v = vv[i] + xv, a = av[i];
    float s = (v >= THETA && a < POSLIM) ? 1.f
            : ((v < -THETA && a > -POSLIM) ? -1.f : 0.f);
    vv[i] = v - s * THETA; a += s; av[i] = a;
    V1[i] = s * THETA;             // v
    V2[i] = (a - s) * THETA;       // v_sum
  }
}

__global__ void k_if_attn(const float* __restrict__ araw, float* va, float* aa,
                          float* __restrict__ C1o, float* __restrict__ C2o) {
  size_t i = (size_t)blockIdx.x * blockDim.x + threadIdx.x;
  if (i >= SZ_NN) return;
  float x = araw[i] * INV_N;
  float v = va[i] + x, a = aa[i];
  float s = (v >= THETA && a < POSLIM) ? 1.f
          : ((v < 0.f && a > 0.f) ? -1.f : 0.f);
  va[i] = v - s * THETA; a += s; aa[i] = a;
  C1o[i] = a * THETA;  // attn + attn_sum
  C2o[i] = s * THETA;  // attn
}

// symmetric IF on y, then [B,H,N,D] -> [B,N,H*D] for the projection GEMM
__global__ void k_if_y(const float* __restrict__ ypre, float* vo, float* ao,
                       float* __restrict__ Y2) {
  size_t i = (size_t)blockIdx.x * blockDim.x + threadIdx.x;
  if (i >= SZ_HD) return;
  int d = (int)(i % D); size_t r = i / D;
  int n = (int)(r % N); r /= N;
  int h = (int)(r % H); int b = (int)(r / H);
  float v = vo[i] + ypre[i], a = ao[i];
  float s = (v >= THETA && a < POSLIM) ? 1.f
          : ((v < -THETA && a > -POSLIM) ? -1.f : 0.f);
  vo[i] = v - s * THETA; a += s; ao[i] = a;
  Y2[((size_t)b * N + n) * C + (size_t)h * D + d] = s * THETA;
}

} // namespace sattn

extern "C" void kernel_launch(void* const* d_in, const int* in_sizes, int n_in,
                              void* d_out, int out_size, void* d_ws,
                              size_t ws_size, hipStream_t stream) {
  using namespace sattn;
  const float* X     = (const float*)d_in[0];
  const float* Wqkv  = (const float*)d_in[1];
  const float* Wproj = (const float*)d_in[2];
  const float* bproj = (const float*)d_in[3];
  float* Out = (float*)d_out;
  float* ws  = (float*)d_ws;
  (void)in_sizes; (void)n_in; (void)out_size; (void)ws_size;

  auto blocksW = [](long waves) { return (int)((waves * 32 + 255) / 256); };

  { // re-init all recurrent state every call (ws is poisoned)
    size_t total = 8 * SZ_HD + 2 * SZ_NN;
    k_init<<<(int)((total + 255) / 256), 256, 0, stream>>>(ws);
  }

  for (int t = 0; t < T; ++t) {
    const float* xt = X + (size_t)t * BN * C;
    float* outt     = Out + (size_t)t * BN * C;

    // qkv = x_t @ w_qkv : [1568,768] @ [768,2304]
    k_gemm_nn<BN, C3, C, C, C3, C3, false>
        <<<blocksW((long)(BN / 32) * (C3 / 32)), 256, 0, stream>>>(
            xt, Wqkv, ws + O_QKV, nullptr);

    k_if_qkv<<<(int)((SZ_HD + 255) / 256), 256, 0, stream>>>(
        ws + O_QKV, ws + O_VQ, ws + O_AQ, ws + O_VK, ws + O_AK,
        ws + O_VV, ws + O_AV, ws + O_A1, ws + O_A2, ws + O_B1, ws + O_B2,
        ws + O_V1, ws + O_V2);

    // attn_raw = A1 @ B1^T + A2 @ B2^T per (b,h)
    k_attn<<<blocksW((long)BH * TN2 * TN2), 256, 0, stream>>>(
        ws + O_A1, ws + O_A2, ws + O_B1, ws + O_B2, ws + O_ATTN);

    k_if_attn<<<(int)((SZ_NN + 255) / 256), 256, 0, stream>>>(
        ws + O_ATTN, ws + O_VA, ws + O_AA, ws + O_C1, ws + O_C2);

    // y_pre = C1 @ v + C2 @ v_sum per (b,h)  (reuses O_ATTN)
    k_yg<<<blocksW((long)BH * TN2 * (D / 32)), 256, 0, stream>>>(
        ws + O_C1, ws + O_C2, ws + O_V1, ws + O_V2, ws + O_ATTN);

    k_if_y<<<(int)((SZ_HD + 255) / 256), 256, 0, stream>>>(
        ws + O_ATTN, ws + O_VO, ws + O_AO, ws + O_Y2);

    // out_t = y2 @ w_proj + b_proj : [1568,768] @ [768,768]
    k_gemm_nn<BN, C, C, C, C, C, true>
        <<<blocksW((long)(BN / 32) * (C / 32)), 256, 0, stream>>>(
            ws + O_Y2, Wproj, outt, bproj);
  }
}